// GATOriginalAttention_78305843741121
// MI455X (gfx1250) — compile-verified
//
#include <hip/hip_runtime.h>
#include <hip/hip_bf16.h>

typedef __attribute__((ext_vector_type(2))) float v2f;
typedef __attribute__((ext_vector_type(4))) float v4f;
typedef __attribute__((ext_vector_type(8))) float v8f;

#define GAT_K 8
#define GAT_D 64

// ---------------------------------------------------------------------------
// Phase 1 (WMMA): el[n,h] = dot(feat_src[n,h,:], attn_l[h,:])
//                 er[n,h] = dot(feat_dst[n,h,:], attn_r[h,:])
// One wave32 per (16-node tile, head). Block = 256 threads = 8 waves = 8 heads.
// V_WMMA_F32_16X16X4_F32 chained 16x over D=64 (K=4 per step).
//   A (16x4 f32): lanes 0-15 hold K=0,1 in v[0],v[1]; lanes 16-31 hold K=2,3.
//   B (4x16 f32): attn chunk replicated across all 16 columns, so every
//   column of the 16x16 accumulator converges to the per-row dot product.
//   C/D (16x16 f32): VGPR r, lanes 0-15 -> row r; lanes 16-31 -> row 8+r.
// ---------------------------------------------------------------------------
__global__ __launch_bounds__(256) void gat_el_er_wmma(
    const float* __restrict__ feat_src, const float* __restrict__ feat_dst,
    const float* __restrict__ attn_l,   const float* __restrict__ attn_r,
    float* __restrict__ el, float* __restrict__ er)
{
    const int h    = threadIdx.x >> 5;      // wave id == head
    const int lane = threadIdx.x & 31;
    const int m    = lane & 15;             // A-matrix row (node within tile)
    const int kh   = lane >> 4;             // 0: K=0,1   1: K=2,3
    const int n0   = blockIdx.x * 16;

    const float* __restrict__ srow = feat_src + ((size_t)(n0 + m) * GAT_K + h) * GAT_D;
    const float* __restrict__ drow = feat_dst + ((size_t)(n0 + m) * GAT_K + h) * GAT_D;
    const float* __restrict__ al   = attn_l + h * GAT_D;
    const float* __restrict__ ar   = attn_r + h * GAT_D;

    v8f cel = {0.f, 0.f, 0.f, 0.f, 0.f, 0.f, 0.f, 0.f};
    v8f cer = {0.f, 0.f, 0.f, 0.f, 0.f, 0.f, 0.f, 0.f};

#pragma unroll
    for (int s = 0; s < GAT_D / 4; ++s) {
        const int off = 4 * s + 2 * kh;     // 8-byte aligned float2 offsets
        v2f a_s = *(const v2f*)(srow + off);
        v2f a_d = *(const v2f*)(drow + off);
        v2f b_l = *(const v2f*)(al + off);  // replicated columns (same in all lanes of a k-row)
        v2f b_r = *(const v2f*)(ar + off);
        // 8 args: (neg_a, A, neg_b, B, c_mod, C, reuse_a, reuse_b)
        cel = __builtin_amdgcn_wmma_f32_16x16x4_f32(false, a_s, false, b_l,
                                                    (short)0, cel, false, false);
        cer = __builtin_amdgcn_wmma_f32_16x16x4_f32(false, a_d, false, b_r,
                                                    (short)0, cer, false, false);
    }

    // Column 0 lives in lane 0 (rows 0-7) and lane 16 (rows 8-15).
    if ((lane & 15) == 0) {
        const int mb = (lane >> 4) * 8;
#pragma unroll
        for (int r = 0; r < 8; ++r) {
            el[(size_t)(n0 + mb + r) * GAT_K + h] = cel[r];
            er[(size_t)(n0 + mb + r) * GAT_K + h] = cer[r];
        }
    }
}

// ---------------------------------------------------------------------------
// Phase 1 tail (scalar) for N % 16 nodes (never fires for N = 100000).
// ---------------------------------------------------------------------------
__global__ void gat_el_er_tail(
    const float* __restrict__ feat_src, const float* __restrict__ feat_dst,
    const float* __restrict__ attn_l,   const float* __restrict__ attn_r,
    float* __restrict__ el, float* __restrict__ er, int nstart, int ncount)
{
    int t = blockIdx.x * blockDim.x + threadIdx.x;
    if (t >= ncount * GAT_K) return;
    int n = nstart + t / GAT_K;
    int h = t % GAT_K;
    const float* ps = feat_src + ((size_t)n * GAT_K + h) * GAT_D;
    const float* pd = feat_dst + ((size_t)n * GAT_K + h) * GAT_D;
    const float* pl = attn_l + h * GAT_D;
    const float* pr = attn_r + h * GAT_D;
    float sl = 0.f, sr = 0.f;
#pragma unroll 8
    for (int d = 0; d < GAT_D; ++d) {
        sl = __builtin_fmaf(ps[d], pl[d], sl);
        sr = __builtin_fmaf(pd[d], pr[d], sr);
    }
    el[(size_t)n * GAT_K + h] = sl;
    er[(size_t)n * GAT_K + h] = sr;
}

// ---------------------------------------------------------------------------
// Phase 2: out[e, 0..7] = el[src_idx[e], 0..7] + er[dst_idx[e], 0..7]
// One thread per edge: coalesced index load, two 32B gathers from the
// L2-resident el/er tables, 32B coalesced store.
// ---------------------------------------------------------------------------
__global__ __launch_bounds__(256) void gat_gather_add(
    const int* __restrict__ src_idx, const int* __restrict__ dst_idx,
    const float* __restrict__ el, const float* __restrict__ er,
    float* __restrict__ out, int E)
{
    int e = blockIdx.x * blockDim.x + threadIdx.x;
    if (e >= E) return;
    int s = src_idx[e];
    int d = dst_idx[e];
    const v4f* pl = (const v4f*)(el + (size_t)s * GAT_K);
    const v4f* pr = (const v4f*)(er + (size_t)d * GAT_K);
    v4f a0 = pl[0], a1 = pl[1];
    v4f b0 = pr[0], b1 = pr[1];
    v4f* po = (v4f*)(out + (size_t)e * GAT_K);
    po[0] = a0 + b0;
    po[1] = a1 + b1;
}

// ---------------------------------------------------------------------------
// Fallback if workspace is too small: recompute dots per output element.
// ---------------------------------------------------------------------------
__global__ void gat_fused_fallback(
    const float* __restrict__ feat_src, const float* __restrict__ feat_dst,
    const float* __restrict__ attn_l,   const float* __restrict__ attn_r,
    const int* __restrict__ src_idx,    const int* __restrict__ dst_idx,
    float* __restrict__ out, long long total)
{
    long long t = (long long)blockIdx.x * blockDim.x + threadIdx.x;
    if (t >= total) return;
    int e = (int)(t >> 3);
    int h = (int)(t & 7);
    int s = src_idx[e];
    int d = dst_idx[e];
    const float* ps = feat_src + ((size_t)s * GAT_K + h) * GAT_D;
    const float* pd = feat_dst + ((size_t)d * GAT_K + h) * GAT_D;
    const float* pl = attn_l + h * GAT_D;
    const float* pr = attn_r + h * GAT_D;
    float acc = 0.f;
#pragma unroll 8
    for (int k = 0; k < GAT_D; ++k)
        acc = __builtin_fmaf(ps[k], pl[k], __builtin_fmaf(pd[k], pr[k], acc));
    out[t] = acc;
}

extern "C" void kernel_launch(void* const* d_in, const int* in_sizes, int n_in,
                              void* d_out, int out_size, void* d_ws, size_t ws_size,
                              hipStream_t stream) {
    const float* feat_src = (const float*)d_in[0];
    const float* feat_dst = (const float*)d_in[1];
    const float* attn_l   = (const float*)d_in[2];
    const float* attn_r   = (const float*)d_in[3];
    const int*   src_idx  = (const int*)d_in[4];
    const int*   dst_idx  = (const int*)d_in[5];
    float* out = (float*)d_out;

    const long long N = (long long)in_sizes[0] / (GAT_K * GAT_D);
    const int E = in_sizes[4];

    const size_t need = (size_t)2 * N * GAT_K * sizeof(float);   // el + er
    if (ws_size >= need) {
        float* el = (float*)d_ws;
        float* er = el + (size_t)N * GAT_K;

        const int ntiles = (int)(N / 16);
        const int nrem   = (int)(N % 16);
        if (ntiles > 0) {
            gat_el_er_wmma<<<ntiles, 256, 0, stream>>>(
                feat_src, feat_dst, attn_l, attn_r, el, er);
        }
        if (nrem > 0) {
            const int thr = nrem * GAT_K;
            gat_el_er_tail<<<(thr + 255) / 256, 256, 0, stream>>>(
                feat_src, feat_dst, attn_l, attn_r, el, er, ntiles * 16, nrem);
        }
        gat_gather_add<<<(E + 255) / 256, 256, 0, stream>>>(
            src_idx, dst_idx, el, er, out, E);
    } else {
        const long long total = (long long)E * GAT_K;
        const long long blocks = (total + 255) / 256;
        gat_fused_fallback<<<(int)blocks, 256, 0, stream>>>(
            feat_src, feat_dst, attn_l, attn_r, src_idx, dst_idx, out, total);
    }
}